// Backbone_66571993088161
// MI455X (gfx1250) — compile-verified
//
#include <hip/hip_runtime.h>
#include <math.h>

// ---------------------------------------------------------------------------
// Shapes (from reference)
#define BATCH   2
#define S_LEN   4096
#define HDIM    1024
#define FDIM    3072
#define GNUM    50
#define MAXK    819          // int(S * 0.2)
#define MAXSPAN 30
#define BANDW   32           // padded band width (30 valid + 2 pad)

typedef float v2f __attribute__((ext_vector_type(2)));
typedef float v8f __attribute__((ext_vector_type(8)));

// D = A(16x4) * B(4x16) + C, all f32, wave32 WMMA
__device__ __forceinline__ v8f wmma_f32(v2f a, v2f b, v8f c) {
    return __builtin_amdgcn_wmma_f32_16x16x4_f32(false, a, false, b,
                                                 (short)0, c, false, false);
}

__device__ __forceinline__ float gelu_erf(float x) {
    return 0.5f * x * (1.0f + erff(x * 0.7071067811865475f));
}

__device__ __forceinline__ unsigned fkey(float f) {
    unsigned u = __float_as_uint(f);
    return (u & 0x80000000u) ? ~u : (u | 0x80000000u);
}

// Async global -> LDS copy of 16 bytes (per lane).  GVS mode:
//   mem = saddr(64b SGPR) + voffset(32b VGPR) ;  lds dest = low 32b of flat ptr
__device__ __forceinline__ void async_copy_b128(void* lds_ptr, const void* base,
                                                int byte_off) {
    unsigned lds = (unsigned)(size_t)lds_ptr;   // LDS aperture: low 32b = offset
    asm volatile("global_load_async_to_lds_b128 %0, %1, %2"
                 :: "v"(lds), "v"(byte_off), "s"(base) : "memory");
}
__device__ __forceinline__ void wait_async_le3() {
    asm volatile("s_wait_asynccnt 0x3" ::: "memory");
}
__device__ __forceinline__ void wait_async_0() {
    asm volatile("s_wait_asynccnt 0x0" ::: "memory");
}

// ---------------------------------------------------------------------------
// GEMM: C[M,N] = act(A[M,K] @ W[K,N] + bias[N]);  act: 0=none, 1=gelu(erf)
// Block: 256 threads = 8 waves (4x2), tile 128x64, K-step 16.
// Double-buffered LDS filled by ASYNCcnt-tracked global_load_async_to_lds_b128
// (3 async instructions / wave / stage), pipelined one stage ahead.
#define BM 128
#define BN 64
#define BK 16
#define APAD 8               // A row stride 24 floats = 96B (16B aligned)

__global__ __launch_bounds__(256)
void gemm_bias_act(const float* __restrict__ A, const float* __restrict__ W,
                   const float* __restrict__ bias, float* __restrict__ C,
                   int M, int N, int K, int do_gelu)
{
    __shared__ float As[2][BM][BK + APAD];
    __shared__ float Bs[2][BK][BN];

    const int tid  = threadIdx.x;
    const int lane = tid & 31;
    const int wave = tid >> 5;
    const int wm   = wave & 3;         // wave row (0..3) -> 32 rows each
    const int wn   = wave >> 2;        // wave col (0..1) -> 32 cols each
    const int mlo  = lane & 15;
    const int khi  = lane >> 4;
    const int bm0  = blockIdx.x * BM;
    const int bn0  = blockIdx.y * BN;

    // Per-stage tile fetch: A 128x16 (512 x b128 -> 2/thread), B 16x64 (1/thread)
    auto issue_stage = [&](int buf, int k0) {
#pragma unroll
        for (int t = 0; t < 2; ++t) {
            const int q  = tid + t * 256;       // 0..511
            const int r  = q >> 2;              // row 0..127
            const int c4 = q & 3;               // 4-col group
            async_copy_b128(&As[buf][r][c4 * 4], A,
                            (int)(((size_t)(bm0 + r) * K + (k0 + c4 * 4)) * 4));
        }
        {
            const int r  = tid >> 4;            // row 0..15
            const int c4 = tid & 15;            // 4-col group
            async_copy_b128(&Bs[buf][r][c4 * 4], W,
                            (int)(((size_t)(k0 + r) * N + (bn0 + c4 * 4)) * 4));
        }
    };

    v8f acc[2][2] = {};

    const int nStages = K / BK;
    issue_stage(0, 0);

    for (int st = 0; st < nStages; ++st) {
        const int buf = st & 1;
        if (st + 1 < nStages) {
            issue_stage(buf ^ 1, (st + 1) * BK);
            wait_async_le3();                  // stage st landed (in-order)
        } else {
            wait_async_0();
        }
        __syncthreads();                       // tile visible to all waves

#pragma unroll
        for (int kk = 0; kk < BK; kk += 4) {
            const int kc = kk + 2 * khi;
            v2f a0, a1, b0, b1;
            // A fragment: lane holds M = mlo, VGPR pair = K {kc, kc+1}
            a0.x = As[buf][wm * 32 +      mlo][kc];
            a0.y = As[buf][wm * 32 +      mlo][kc + 1];
            a1.x = As[buf][wm * 32 + 16 + mlo][kc];
            a1.y = As[buf][wm * 32 + 16 + mlo][kc + 1];
            // B fragment: lane holds N = mlo, VGPR pair = K {kc, kc+1}
            b0.x = Bs[buf][kc    ][wn * 32 +      mlo];
            b0.y = Bs[buf][kc + 1][wn * 32 +      mlo];
            b1.x = Bs[buf][kc    ][wn * 32 + 16 + mlo];
            b1.y = Bs[buf][kc + 1][wn * 32 + 16 + mlo];

            acc[0][0] = wmma_f32(a0, b0, acc[0][0]);
            acc[0][1] = wmma_f32(a0, b1, acc[0][1]);
            acc[1][0] = wmma_f32(a1, b0, acc[1][0]);
            acc[1][1] = wmma_f32(a1, b1, acc[1][1]);
        }
        __syncthreads();                       // retire buf before overwrite
    }

    // C layout: VGPR r -> (M = r + 8*khi, N = mlo) within each 16x16 tile
#pragma unroll
    for (int ti = 0; ti < 2; ++ti)
#pragma unroll
        for (int tj = 0; tj < 2; ++tj) {
            const int col = bn0 + wn * 32 + tj * 16 + mlo;
            const float bv = bias[col];
            const int rb = bm0 + wm * 32 + ti * 16 + 8 * khi;
#pragma unroll
            for (int r = 0; r < 8; ++r) {
                float v = acc[ti][tj][r] + bv;
                if (do_gelu) v = gelu_erf(v);
                C[(size_t)(rb + r) * N + col] = v;
            }
        }
}

// ---------------------------------------------------------------------------
// Per-row LayerNorm over F, in place: h = (h-mu)*rsqrt(var+eps)*g + beta
__global__ __launch_bounds__(256)
void layernorm_rows(float* __restrict__ h, const float* __restrict__ g,
                    const float* __restrict__ beta)
{
    __shared__ float r1[256], r2[256];
    const int tid = threadIdx.x;
    float* row = h + (size_t)blockIdx.x * FDIM;
    float s = 0.f, sq = 0.f;
    for (int f = tid; f < FDIM; f += 256) { float v = row[f]; s += v; sq += v * v; }
    r1[tid] = s; r2[tid] = sq; __syncthreads();
    for (int off = 128; off > 0; off >>= 1) {
        if (tid < off) { r1[tid] += r1[tid + off]; r2[tid] += r2[tid + off]; }
        __syncthreads();
    }
    const float mu = r1[0] * (1.0f / FDIM);
    const float var = r2[0] * (1.0f / FDIM) - mu * mu;
    const float rs = rsqrtf(var + 1e-5f);
    for (int f = tid; f < FDIM; f += 256)
        row[f] = (row[f] - mu) * rs * g[f] + beta[f];
}

// ---------------------------------------------------------------------------
// out[row] = reps[row,:] . w + b
__global__ __launch_bounds__(256)
void row_dot(const float* __restrict__ reps, const float* __restrict__ w,
             const float* __restrict__ b, float* __restrict__ out)
{
    __shared__ float red[256];
    const int tid = threadIdx.x;
    const float* row = reps + (size_t)blockIdx.x * FDIM;
    float s = 0.f;
    for (int f = tid; f < FDIM; f += 256) s += row[f] * w[f];
    red[tid] = s; __syncthreads();
    for (int off = 128; off > 0; off >>= 1) {
        if (tid < off) red[tid] += red[tid + off];
        __syncthreads();
    }
    if (tid == 0) out[blockIdx.x] = red[0] + b[0];
}

// ---------------------------------------------------------------------------
__global__ void init_ws(float* __restrict__ band, float* __restrict__ partJ,
                        float* __restrict__ partM)
{
    const int i = blockIdx.x * 256 + threadIdx.x;
    if (i < BATCH * S_LEN * BANDW) band[i] = -__builtin_inff();
    if (i < BATCH * 256) { partJ[i] = 0.f; partM[i] = 0.f; }
}

// ---------------------------------------------------------------------------
// Banded bilinear:  logits(s,e) = temp[b,s,:] . end_reps[b,e,:] + sl[s] + el[e]
// for 0 <= e-s < 30.  One block per (16-row stripe, batch); 3 waves cover the
// 16x48 diagonal tile via WMMA over K=3072.  Fused: clip, store band value,
// and accumulate BCE(t=0) + mask-count partial sums.
__global__ __launch_bounds__(96)
void band_kernel(const float* __restrict__ tempM, const float* __restrict__ endReps,
                 const float* __restrict__ sl, const float* __restrict__ el,
                 const int* __restrict__ attn,
                 float* __restrict__ band, float* __restrict__ partJ,
                 float* __restrict__ partM)
{
    __shared__ float redJ[96], redM[96];
    const int tid  = threadIdx.x;
    const int lane = tid & 31;
    const int wave = tid >> 5;          // 0..2 -> e-tile
    const int mlo  = lane & 15;
    const int khi  = lane >> 4;
    const int b    = blockIdx.y;
    const int s0   = blockIdx.x * 16;

    const int e_lane = s0 + wave * 16 + mlo;
    const int e_c    = e_lane < S_LEN ? e_lane : (S_LEN - 1);
    const float* Arow = tempM   + (size_t)(b * S_LEN + s0 + mlo) * FDIM;
    const float* Brow = endReps + (size_t)(b * S_LEN + e_c)      * FDIM;

    v8f acc = {};
    for (int k = 0; k < FDIM; k += 4) {
        v2f a, bb;
        a.x  = Arow[k + 2 * khi];  a.y  = Arow[k + 2 * khi + 1];
        bb.x = Brow[k + 2 * khi];  bb.y = Brow[k + 2 * khi + 1];
        acc = wmma_f32(a, bb, acc);
    }

    float lj = 0.f, lm = 0.f;
#pragma unroll
    for (int r = 0; r < 8; ++r) {
        const int s = s0 + r + 8 * khi;
        const int e = e_lane;
        const int d = e - s;
        if (d >= 0 && d < MAXSPAN && e < S_LEN) {
            float x = acc[r] + sl[b * S_LEN + s] + el[b * S_LEN + e];
            x = fminf(fmaxf(x, -10000.f), 10000.f);
            band[(size_t)(b * S_LEN + s) * BANDW + d] = x;
            const float w = (float)(attn[b * S_LEN + s] * attn[b * S_LEN + e]);
            if (w != 0.f) {
                const float p = 1.f / (1.f + expf(-x));
                lj += w * (-fmaxf(log1pf(-p), -100.f));   // bce(p, 0)
                lm += w;
            }
        }
    }
    redJ[tid] = lj; redM[tid] = lm; __syncthreads();
    if (tid == 0) {
        float tj = 0.f, tm = 0.f;
        for (int i = 0; i < 96; ++i) { tj += redJ[i]; tm += redM[i]; }
        partJ[b * 256 + blockIdx.x] = tj;
        partM[b * 256 + blockIdx.x] = tm;
    }
}

// ---------------------------------------------------------------------------
// Gold BCE + final cost.  One block per batch.
__global__ __launch_bounds__(256)
void final_kernel(const float* __restrict__ tempM, const float* __restrict__ endReps,
                  const float* __restrict__ sl, const float* __restrict__ el,
                  const int* __restrict__ attn, const int* __restrict__ gold,
                  const float* __restrict__ goldMask,
                  const float* __restrict__ partJ, const float* __restrict__ partM,
                  float* __restrict__ outCost)
{
    __shared__ float red[256];
    __shared__ int gs_s[GNUM], ge_s[GNUM], dup_s[GNUM];
    __shared__ int sh_hasGold;
    __shared__ float sh_totJ, sh_totM;
    const int tid = threadIdx.x;
    const int b = blockIdx.x;

    red[tid] = partJ[b * 256 + tid]; __syncthreads();
    for (int off = 128; off > 0; off >>= 1) {
        if (tid < off) red[tid] += red[tid + off];
        __syncthreads();
    }
    if (tid == 0) sh_totJ = red[0];
    __syncthreads();
    red[tid] = partM[b * 256 + tid]; __syncthreads();
    for (int off = 128; off > 0; off >>= 1) {
        if (tid < off) red[tid] += red[tid + off];
        __syncthreads();
    }
    if (tid == 0) sh_totM = red[0];

    if (tid < GNUM) {
        gs_s[tid] = gold[(b * GNUM + tid) * 2 + 0];
        ge_s[tid] = gold[(b * GNUM + tid) * 2 + 1];
    }
    __syncthreads();
    if (tid == 0) {
        int hs = 0;
        for (int g = 0; g < GNUM; ++g) {
            hs += gs_s[g] + ge_s[g];
            int dp = 0;
            for (int h = 0; h < g; ++h)
                if (gs_s[h] == gs_s[g] && ge_s[h] == ge_s[g]) { dp = 1; break; }
            dup_s[g] = dp;
        }
        sh_hasGold = (hs > 0);
    }
    __syncthreads();

    float costGold = 0.f, corr = 0.f, gmSum = 0.f;   // thread 0 accumulates
    for (int g = 0; g < GNUM; ++g) {
        const float* ar = tempM   + (size_t)(b * S_LEN + gs_s[g]) * FDIM;
        const float* br = endReps + (size_t)(b * S_LEN + ge_s[g]) * FDIM;
        float s = 0.f;
        for (int f = tid; f < FDIM; f += 256) s += ar[f] * br[f];
        red[tid] = s; __syncthreads();
        for (int off = 128; off > 0; off >>= 1) {
            if (tid < off) red[tid] += red[tid + off];
            __syncthreads();
        }
        if (tid == 0) {
            const int gsv = gs_s[g], gev = ge_s[g];
            float x = red[0] + sl[b * S_LEN + gsv] + el[b * S_LEN + gev];
            const int valid = (gev >= gsv) && (gev < gsv + MAXSPAN);
            if (!valid) x -= 10000.f;
            x = fminf(fmaxf(x, -10000.f), 10000.f);
            const float p = 1.f / (1.f + expf(-x));
            const float gm = goldMask[b * GNUM + g];
            gmSum += gm;
            const float pe  = p * gm;
            const float lp  = fmaxf(logf(pe), -100.f);
            const float l1p = fmaxf(log1pf(-pe), -100.f);
            costGold += -(gm * lp + (1.f - gm) * l1p);   // bce(p*gm, gm)
            // junk zeroing correction (skip (0,0): it is restored)
            if (sh_hasGold && valid && !dup_s[g] && !(gsv == 0 && gev == 0)) {
                if (attn[b * S_LEN + gsv] * attn[b * S_LEN + gev] != 0)
                    corr += -fmaxf(log1pf(-p), -100.f);
            }
        }
        __syncthreads();
    }
    if (tid == 0) {
        const float costJunk = (sh_totJ - corr) / sh_totM;
        const float cg = sh_hasGold ? (costGold / gmSum) : 0.f;
        outCost[b] = cg + costJunk;
    }
}

// ---------------------------------------------------------------------------
// Top-k over the band (the top-819 provably lies in-band) via 4-round 8-bit
// radix select on monotone keys, then bitonic index sort.  One block / batch.
__device__ __forceinline__ void bitonic1024(int* buf, int tid)
{
    for (int k = 2; k <= 1024; k <<= 1)
        for (int j = k >> 1; j > 0; j >>= 1) {
            const int ix = tid ^ j;
            if (ix > tid) {
                const bool up = ((tid & k) == 0);
                const int a = buf[tid], c = buf[ix];
                if ((a > c) == up) { buf[tid] = c; buf[ix] = a; }
            }
            __syncthreads();
        }
}

__global__ __launch_bounds__(1024)
void topk_kernel(const float* __restrict__ band, const int* __restrict__ attn,
                 float* __restrict__ outStarts, float* __restrict__ outEnds,
                 float* __restrict__ outMask)
{
    __shared__ unsigned hist[256];
    __shared__ int arr[1024];
    __shared__ int eqArr[1024];
    __shared__ int red_i[1024];
    __shared__ int sh_digit, sh_cum, sh_gtc, sh_eqc;

    const int tid = threadIdx.x;
    const int b = blockIdx.x;
    const int N = S_LEN * BANDW;
    const float* bb = band + (size_t)b * N;

    int ls = 0;
    for (int i = tid; i < S_LEN; i += 1024) ls += attn[b * S_LEN + i];
    red_i[tid] = ls; __syncthreads();
    for (int off = 512; off > 0; off >>= 1) {
        if (tid < off) red_i[tid] += red_i[tid + off];
        __syncthreads();
    }
    int kk = (int)((float)red_i[0] * 0.2f);
    if (kk > MAXK) kk = MAXK;

    unsigned prefix = 0u, pmask = 0u;
    int need = kk;
    for (int round = 0; round < 4; ++round) {
        const int shift = 24 - 8 * round;
        if (tid < 256) hist[tid] = 0u;
        __syncthreads();
        for (int i = tid; i < N; i += 1024) {
            const unsigned key = fkey(bb[i]);
            if ((key & pmask) == prefix)
                atomicAdd(&hist[(key >> shift) & 255], 1u);
        }
        __syncthreads();
        if (tid == 0) {
            int cum = 0, dig = 0;
            for (int bk = 255; bk >= 0; --bk) {
                const int h = (int)hist[bk];
                if (cum + h >= need) { dig = bk; break; }
                cum += h;
            }
            sh_digit = dig; sh_cum = cum;
        }
        __syncthreads();
        need  -= sh_cum;
        prefix |= ((unsigned)sh_digit) << shift;
        pmask  |= (0xFFu << shift);
        __syncthreads();
    }
    const unsigned T = prefix;
    const int needEq = need;

    if (tid == 0) { sh_gtc = 0; sh_eqc = 0; }
    __syncthreads();
    for (int i = tid; i < N; i += 1024) {
        const unsigned key = fkey(bb[i]);
        const int s = i >> 5;
        const int flat = s * S_LEN + s + (i & 31);   // s*S + e
        if (key > T) {
            const int p = atomicAdd(&sh_gtc, 1);
            if (p < 1024) arr[p] = flat;
        } else if (key == T) {
            const int p = atomicAdd(&sh_eqc, 1);
            if (p < 1024) eqArr[p] = flat;
        }
    }
    __syncthreads();
    int gtc = sh_gtc; if (gtc > 1024) gtc = 1024;
    int eqc = sh_eqc; if (eqc > 1024) eqc = 1024;

    if (tid >= eqc) eqArr[tid] = 0x7FFFFFFF;
    __syncthreads();
    bitonic1024(eqArr, tid);             // jax ties: smallest index first

    const int comb = (tid < gtc) ? arr[tid]
                   : ((tid < gtc + needEq) ? eqArr[tid - gtc] : 0x7FFFFFFF);
    __syncthreads();
    arr[tid] = comb;
    __syncthreads();
    bitonic1024(arr, tid);               // ascending index order (jnp.sort)

    if (tid < MAXK) {
        int idx = 0; float mv = 0.f;
        if (tid < kk) {
            mv = 1.f;
            idx = arr[tid];
            if (idx == S_LEN * S_LEN - 1 || idx == 0x7FFFFFFF) idx = 0;
        }
        outStarts[b * MAXK + tid] = (float)(idx / S_LEN);
        outEnds  [b * MAXK + tid] = (float)(idx % S_LEN);
        outMask  [b * MAXK + tid] = mv;
    }
}

// ---------------------------------------------------------------------------
extern "C" void kernel_launch(void* const* d_in, const int* in_sizes, int n_in,
                              void* d_out, int out_size, void* d_ws, size_t ws_size,
                              hipStream_t stream)
{
    (void)in_sizes; (void)n_in; (void)out_size; (void)ws_size;
    const float* seq   = (const float*)d_in[0];
    const int*   attn  = (const int*)d_in[1];
    const int*   gold  = (const int*)d_in[2];
    const float* gmask = (const float*)d_in[3];
    const float* Ws    = (const float*)d_in[4];
    const float* bs    = (const float*)d_in[5];
    const float* gs    = (const float*)d_in[6];
    const float* betas = (const float*)d_in[7];
    const float* We    = (const float*)d_in[8];
    const float* be    = (const float*)d_in[9];
    const float* geV   = (const float*)d_in[10];
    const float* betae = (const float*)d_in[11];
    const float* wst   = (const float*)d_in[12];
    const float* bst   = (const float*)d_in[13];
    const float* wen   = (const float*)d_in[14];
    const float* ben   = (const float*)d_in[15];
    const float* Ws2e  = (const float*)d_in[16];
    const float* bs2e  = (const float*)d_in[17];
    float* out = (float*)d_out;

    const int M = BATCH * S_LEN;                       // 8192

    // workspace layout (floats)
    float* ws        = (float*)d_ws;
    float* startReps = ws;                             // 8192*3072
    float* endReps   = startReps + (size_t)M * FDIM;
    float* tempBuf   = endReps   + (size_t)M * FDIM;
    float* sl        = tempBuf   + (size_t)M * FDIM;   // 8192
    float* el        = sl + M;                         // 8192
    float* band      = el + M;                         // 2*4096*32
    float* partJ     = band + (size_t)BATCH * S_LEN * BANDW;
    float* partM     = partJ + BATCH * 256;

    // output layout (tuple flattened): starts, ends, span_mask, seq, cost
    float* oStarts = out;
    float* oEnds   = out + BATCH * MAXK;
    float* oMask   = out + 2 * BATCH * MAXK;
    float* oSeq    = out + 3 * BATCH * MAXK;
    float* oCost   = oSeq + (size_t)BATCH * S_LEN * HDIM;

    hipMemcpyAsync(oSeq, seq, (size_t)BATCH * S_LEN * HDIM * sizeof(float),
                   hipMemcpyDeviceToDevice, stream);

    init_ws<<<(BATCH * S_LEN * BANDW + 255) / 256, 256, 0, stream>>>(band, partJ, partM);

    dim3 g1(M / BM, FDIM / BN);
    gemm_bias_act<<<g1, 256, 0, stream>>>(seq, Ws, bs, startReps, M, FDIM, HDIM, 1);
    gemm_bias_act<<<g1, 256, 0, stream>>>(seq, We, be, endReps,  M, FDIM, HDIM, 1);

    layernorm_rows<<<M, 256, 0, stream>>>(startReps, gs,  betas);
    layernorm_rows<<<M, 256, 0, stream>>>(endReps,  geV, betae);

    row_dot<<<M, 256, 0, stream>>>(startReps, wst, bst, sl);
    row_dot<<<M, 256, 0, stream>>>(endReps,  wen, ben, el);

    gemm_bias_act<<<g1, 256, 0, stream>>>(startReps, Ws2e, bs2e, tempBuf, M, FDIM, FDIM, 0);

    dim3 g2(S_LEN / 16, BATCH);
    band_kernel<<<g2, 96, 0, stream>>>(tempBuf, endReps, sl, el, attn,
                                       band, partJ, partM);

    final_kernel<<<BATCH, 256, 0, stream>>>(tempBuf, endReps, sl, el, attn,
                                            gold, gmask, partJ, partM, oCost);

    topk_kernel<<<BATCH, 1024, 0, stream>>>(band, attn, oStarts, oEnds, oMask);
}